// CellMemoryGraph_25280177504281
// MI455X (gfx1250) — compile-verified
//
#include <hip/hip_runtime.h>
#include <hip/hip_bf16.h>
#include <math.h>

// ---- problem constants (from reference) ----
#define BSZ     8
#define NCC     64
#define CC      256
#define DD      32
#define KK      16
#define KBB     8
#define AALPHA  4
#define BB      16       // border cells per nc
#define HSs     64
#define HMm     64
#define HMODh   32
#define NTOT    (NCC*CC)         // 16384
#define MOD_IN   113
#define MOD_INP  116             // padded to multiple of 4
#define MOD_OUT  57
#define MOD_OUTP 64              // padded to multiple of 16

typedef __attribute__((ext_vector_type(2))) float v2f;
typedef __attribute__((ext_vector_type(8))) float v8f;

__device__ __forceinline__ v8f wmma4(v2f a, v2f b, v8f c) {
  // V_WMMA_F32_16X16X4_F32 : D(16x16 f32) = A(16x4 f32) * B(4x16 f32) + C
  return __builtin_amdgcn_wmma_f32_16x16x4_f32(false, a, false, b, (short)0, c,
                                               false, false);
}

__device__ __forceinline__ float sigm(float x) { return 1.0f / (1.0f + expf(-x)); }

// =====================================================================
// Kernel 1: per-neuron modulation MLP.  One wave (32 threads) per neuron.
//   hid[b,h] = tanh( sum_i X[b,i]*W1[n,h,i] + b1[n,h] )   (113 -> 32)
//   out[b,o] =       sum_h hid[b,h]*W2[n,h,o] + b2[n,o]   (32 -> 57)
// WMMA orientation: A = weights (M=rows), B = X^T (N=batch, 8 of 16 used).
// All operand tiles are zero-padded in LDS so fragment loads are
// unconditional, 8B-aligned ds_load_b64 (no exec-mask juggling).
// =====================================================================
__global__ __launch_bounds__(32)
void k_mod_mlp(const float* __restrict__ heb, const float* __restrict__ h,
               const float* __restrict__ dlog, const float* __restrict__ prim,
               const float* __restrict__ nid,
               const float* __restrict__ w1, const float* __restrict__ b1,
               const float* __restrict__ w2, const float* __restrict__ b2,
               float* __restrict__ mod_out)
{
  __shared__ float W1s[HMODh * MOD_INP];     // [h][i]  32 x 116 (zero-padded K)
  __shared__ float Xs [16 * MOD_INP];        // [b][i]  16 x 116 (rows 8..15 zero)
  __shared__ float W2T[MOD_OUTP * HMODh];    // [o][h]  64 x 32  (rows 57..63 zero)
  __shared__ float HIDT[16 * HMODh];         // [b][h]  16 x 32  (rows 8..15 zero)

  const int n    = blockIdx.x;               // neuron 0..16383
  const int lane = threadIdx.x;
  const int l16  = lane & 15;
  const int half = lane >> 4;
  const int nc   = n >> 8;
  const int c    = n & 255;

  // stage W1 (padded) — source rows are contiguous per neuron
  for (int idx = lane; idx < HMODh * MOD_INP; idx += 32) {
    int hh = idx / MOD_INP, i = idx % MOD_INP;
    W1s[idx] = (i < MOD_IN) ? w1[(size_t)n * HMODh * MOD_IN + hh * MOD_IN + i] : 0.0f;
  }
  // stage W2 transposed: W2T[o][h] = w2[n][h][o]; coalesced global reads
  for (int idx = lane; idx < HMODh * MOD_OUT; idx += 32) {
    int hh = idx / MOD_OUT, oo = idx % MOD_OUT;
    W2T[oo * HMODh + hh] = w2[(size_t)n * HMODh * MOD_OUT + idx];
  }
  for (int idx = lane; idx < (MOD_OUTP - MOD_OUT) * HMODh; idx += 32)
    W2T[MOD_OUT * HMODh + idx] = 0.0f;
  // build X[b][i] = concat(hebbian16, h32, decay1, prim32, nid32); pad rows+K
  for (int idx = lane; idx < 16 * MOD_INP; idx += 32) {
    int bb = idx / MOD_INP, i = idx % MOD_INP;
    float v = 0.0f;
    if (bb < BSZ && i < MOD_IN) {
      size_t cf = ((size_t)bb * NCC + nc) * CC + c;
      if      (i < KK)            v = heb[cf * KK + i];
      else if (i < KK + DD)       v = h[cf * DD + (i - KK)];
      else if (i == KK + DD)      v = dlog[cf];
      else if (i < KK + 2*DD + 1) v = prim[cf * DD + (i - (KK + DD + 1))];
      else                        v = nid[((size_t)nc * CC + c) * DD + (i - (KK + 2*DD + 1))];
    }
    Xs[idx] = v;
  }
  __syncthreads();

  // ---- layer 1: 2 M-tiles (h 0..15, 16..31), K loop over 116 ----
  v8f acc0 = 0.0f, acc1 = 0.0f;
  for (int k0 = 0; k0 < MOD_INP; k0 += 4) {
    int kk = k0 + 2 * half;
    v2f bfr;
    bfr.x = Xs[l16 * MOD_INP + kk];
    bfr.y = Xs[l16 * MOD_INP + kk + 1];
    v2f a0, a1;
    a0.x = W1s[l16 * MOD_INP + kk];          a0.y = W1s[l16 * MOD_INP + kk + 1];
    a1.x = W1s[(16 + l16) * MOD_INP + kk];   a1.y = W1s[(16 + l16) * MOD_INP + kk + 1];
    acc0 = wmma4(a0, bfr, acc0);
    acc1 = wmma4(a1, bfr, acc1);
  }
  // hid -> LDS transposed [b][h]  (D layout: col b = l16, row h = 16t + r + 8*half)
#pragma unroll
  for (int r = 0; r < 8; ++r) {
    int h0 = r + 8 * half;
    int h1 = 16 + r + 8 * half;
    float v0 = tanhf(acc0[r] + b1[(size_t)n * HMODh + h0]);
    float v1 = tanhf(acc1[r] + b1[(size_t)n * HMODh + h1]);
    HIDT[l16 * HMODh + h0] = (l16 < BSZ) ? v0 : 0.0f;
    HIDT[l16 * HMODh + h1] = (l16 < BSZ) ? v1 : 0.0f;
  }
  __syncthreads();

  // ---- layer 2: 4 M-tiles over o (57 padded to 64), K loop over h=32 ----
  v8f acc[4];
#pragma unroll
  for (int t = 0; t < 4; ++t) acc[t] = 0.0f;
  for (int h0 = 0; h0 < HMODh; h0 += 4) {
    int kk = h0 + 2 * half;
    v2f bfr;
    bfr.x = HIDT[l16 * HMODh + kk];
    bfr.y = HIDT[l16 * HMODh + kk + 1];
#pragma unroll
    for (int t = 0; t < 4; ++t) {
      int oo = 16 * t + l16;
      v2f a;
      a.x = W2T[oo * HMODh + kk];        // A[o][h] = W2[h][o]
      a.y = W2T[oo * HMODh + kk + 1];
      acc[t] = wmma4(a, bfr, acc[t]);
    }
  }
  if (l16 < BSZ) {
#pragma unroll
    for (int t = 0; t < 4; ++t)
#pragma unroll
      for (int r = 0; r < 8; ++r) {
        int oo = 16 * t + r + 8 * half;
        if (oo < MOD_OUT)
          mod_out[((size_t)l16 * NTOT + n) * MOD_OUT + oo] =
              acc[t][r] + b2[(size_t)n * MOD_OUT + oo];
      }
  }
}

// =====================================================================
// Kernel 2: received = sigmoid-weighted neighbor sums + border + injection
// One wave per (b, nc, c); lane = d.
// =====================================================================
__global__ __launch_bounds__(128)
void k_received(const float* __restrict__ pmsg, const float* __restrict__ mod_out,
                const int* __restrict__ conn, const int* __restrict__ bconn,
                const float* __restrict__ cc, float* __restrict__ recv)
{
  const int wid = threadIdx.x >> 5, lane = threadIdx.x & 31;
  const long cell = (long)blockIdx.x * 4 + wid;     // 0..131071
  const int b  = (int)(cell / NTOT);
  const int n  = (int)(cell % NTOT);
  const int nc = n >> 8, c = n & 255;
  const float* wrow = mod_out + ((long)b * NTOT + n) * MOD_OUT;
  const float* pmb  = pmsg + (long)b * NTOT * DD;

  float acc = 0.0f;
#pragma unroll
  for (int k = 0; k < KK; ++k) {
    int idx = conn[((long)nc * CC + c) * KK + k];
    acc += sigm(wrow[k]) * pmb[((long)nc * CC + idx) * DD + lane];
  }
  if (c >= AALPHA && c < AALPHA + BB) {
    int j = c - AALPHA;
#pragma unroll
    for (int kb = 0; kb < KBB; ++kb) {
      int bi  = bconn[((long)nc * BB + j) * KBB + kb];
      int nc2 = bi >> 4, jj = bi & 15;
      acc += sigm(wrow[KK + kb]) * pmb[((long)nc2 * CC + (AALPHA + jj)) * DD + lane];
    }
  }
  if (c < AALPHA) acc += cc[(long)b * (NCC * DD) + nc * DD + lane];
  recv[cell * DD + lane] = acc;
}

// =====================================================================
// Kernel 3: fused state MLP (97->64->32) + decay mix + msg MLP (96->64->32).
// One wave per tile of 16 consecutive cells; WMMA M=cells, N=hidden, K=feat.
// Shared weights read straight from global (tiny, cache-resident).
// K-loops run unguarded; the single ragged state-layer-1 iteration (k=96)
// is peeled with a branchless select.
// =====================================================================
__global__ __launch_bounds__(128)
void k_state_msg(const float* __restrict__ h_in, const float* __restrict__ recv,
                 const float* __restrict__ prim, const float* __restrict__ nid,
                 const float* __restrict__ mod_out,
                 const float* __restrict__ sw1, const float* __restrict__ sb1,
                 const float* __restrict__ sw2, const float* __restrict__ sb2,
                 const float* __restrict__ mw1, const float* __restrict__ mb1,
                 const float* __restrict__ mw2, const float* __restrict__ mb2,
                 float* __restrict__ h_new_out, float* __restrict__ msg_out)
{
  __shared__ float SIN[4][16 * 100];   // per-wave s_in / m_in tile (K padded to 100)
  __shared__ float SH [4][16 * 64];    // per-wave hidden tile

  const int wid  = threadIdx.x >> 5;
  const int lane = threadIdx.x & 31;
  const int l16  = lane & 15;
  const int half = lane >> 4;
  const long tile  = (long)blockIdx.x * 4 + wid;   // 0..8191
  const long flat0 = tile * 16;                    // flat cell = (b*NC+nc)*C + c
  const int  b     = (int)(flat0 / NTOT);
  const int  n0    = (int)(flat0 % NTOT);          // neuron index of cell 0 of tile
  float* sin_ = SIN[wid];
  float* sh_  = SH[wid];

  // ---- build s_in = [h(32) | received(32) | prim+new_prim(32) | decay(1) | pad] ----
  for (int idx = lane; idx < 16 * 100; idx += 32) {
    int cell = idx / 100, i = idx % 100;
    long cf = flat0 + cell;
    long nn = (long)n0 + cell;
    float v;
    if      (i < DD)     v = h_in[cf * DD + i];
    else if (i < 2*DD)   v = recv[cf * DD + (i - DD)];
    else if (i < 3*DD)   v = prim[cf * DD + (i - 2*DD)] +
                             mod_out[((long)b * NTOT + nn) * MOD_OUT + (KK + KBB + 1) + (i - 2*DD)];
    else if (i == 96)    v = sigm(mod_out[((long)b * NTOT + nn) * MOD_OUT + (KK + KBB)]);
    else                 v = 0.0f;
    sin_[idx] = v;
  }
  __syncthreads();

  // ---- state layer 1: (16 x 97) @ (97 x 64); 24 clean iters + peeled k=96 ----
  v8f acc[4];
#pragma unroll
  for (int t = 0; t < 4; ++t) acc[t] = 0.0f;
  for (int k0 = 0; k0 < 96; k0 += 4) {
    int kk = k0 + 2 * half;
    v2f a;  a.x = sin_[l16 * 100 + kk];  a.y = sin_[l16 * 100 + kk + 1];
#pragma unroll
    for (int t = 0; t < 4; ++t) {
      int nh = 16 * t + l16;
      v2f bf;  bf.x = sw1[nh * 97 + kk];  bf.y = sw1[nh * 97 + kk + 1];
      acc[t] = wmma4(a, bf, acc[t]);
    }
  }
  {
    int kk = 96 + 2 * half;
    v2f a;  a.x = sin_[l16 * 100 + kk];  a.y = sin_[l16 * 100 + kk + 1];
#pragma unroll
    for (int t = 0; t < 4; ++t) {
      int nh = 16 * t + l16;
      float w = sw1[nh * 97 + 96];              // unconditional, valid address
      v2f bf;  bf.x = half ? 0.0f : w;  bf.y = 0.0f;
      acc[t] = wmma4(a, bf, acc[t]);
    }
  }
#pragma unroll
  for (int t = 0; t < 4; ++t)
#pragma unroll
    for (int r = 0; r < 8; ++r) {
      int cell = r + 8 * half;
      int nh   = 16 * t + l16;
      sh_[cell * 64 + nh] = tanhf(acc[t][r] + sb1[nh]);
    }
  __syncthreads();

  // ---- state layer 2: (16 x 64) @ (64 x 32), fused decay mix ----
  v8f acc2[2];  acc2[0] = 0.0f;  acc2[1] = 0.0f;
  for (int h0 = 0; h0 < 64; h0 += 4) {
    int kk = h0 + 2 * half;
    v2f a;  a.x = sh_[l16 * 64 + kk];  a.y = sh_[l16 * 64 + kk + 1];
#pragma unroll
    for (int t = 0; t < 2; ++t) {
      int nd = 16 * t + l16;
      v2f bf;  bf.x = sw2[nd * 64 + kk];  bf.y = sw2[nd * 64 + kk + 1];
      acc2[t] = wmma4(a, bf, acc2[t]);
    }
  }
#pragma unroll
  for (int t = 0; t < 2; ++t)
#pragma unroll
    for (int r = 0; r < 8; ++r) {
      int cell = r + 8 * half;
      int dd   = 16 * t + l16;
      long cf  = flat0 + cell;
      float decay = sin_[cell * 100 + 96];
      float hv = h_in[cf * DD + dd];
      float hn = decay * hv + (1.0f - decay) * tanhf(acc2[t][r] + sb2[dd]);
      h_new_out[cf * DD + dd] = hn;
      sin_[cell * 100 + dd] = hn;           // m_in[0..31] = h_new
    }
  __syncthreads();

  // ---- rebuild m_in tail: [32..63]=received (kept), [64..95]=nid, [96..99]=0 ----
  for (int idx = lane; idx < 16 * 36; idx += 32) {
    int cell = idx / 36, i = idx % 36;
    long cf = flat0 + cell;
    int ncc = (int)(cf % NTOT);             // nc*C + c
    sin_[cell * 100 + 64 + i] = (i < 32) ? nid[(long)ncc * DD + i] : 0.0f;
  }
  __syncthreads();

  // ---- msg layer 1: (16 x 96) @ (96 x 64); 24 exact iterations ----
#pragma unroll
  for (int t = 0; t < 4; ++t) acc[t] = 0.0f;
  for (int k0 = 0; k0 < 96; k0 += 4) {
    int kk = k0 + 2 * half;
    v2f a;  a.x = sin_[l16 * 100 + kk];  a.y = sin_[l16 * 100 + kk + 1];
#pragma unroll
    for (int t = 0; t < 4; ++t) {
      int nh = 16 * t + l16;
      v2f bf;  bf.x = mw1[nh * 96 + kk];  bf.y = mw1[nh * 96 + kk + 1];
      acc[t] = wmma4(a, bf, acc[t]);
    }
  }
#pragma unroll
  for (int t = 0; t < 4; ++t)
#pragma unroll
    for (int r = 0; r < 8; ++r) {
      int cell = r + 8 * half;
      int nh   = 16 * t + l16;
      sh_[cell * 64 + nh] = tanhf(acc[t][r] + mb1[nh]);
    }
  __syncthreads();

  // ---- msg layer 2: (16 x 64) @ (64 x 32) ----
  acc2[0] = 0.0f;  acc2[1] = 0.0f;
  for (int h0 = 0; h0 < 64; h0 += 4) {
    int kk = h0 + 2 * half;
    v2f a;  a.x = sh_[l16 * 64 + kk];  a.y = sh_[l16 * 64 + kk + 1];
#pragma unroll
    for (int t = 0; t < 2; ++t) {
      int nd = 16 * t + l16;
      v2f bf;  bf.x = mw2[nd * 64 + kk];  bf.y = mw2[nd * 64 + kk + 1];
      acc2[t] = wmma4(a, bf, acc2[t]);
    }
  }
#pragma unroll
  for (int t = 0; t < 2; ++t)
#pragma unroll
    for (int r = 0; r < 8; ++r) {
      int cell = r + 8 * half;
      int dd   = 16 * t + l16;
      long cf  = flat0 + cell;
      msg_out[cf * DD + dd] = tanhf(acc2[t][r] + mb2[dd]);
    }
}

// =====================================================================
// Kernel 4: hebbian_new = 0.9*heb + 0.1 * dot(msg, msg[neighbor]) / D
// One wave per cell; full wave32 shuffle reduction per k.
// =====================================================================
__global__ __launch_bounds__(128)
void k_hebbian(const float* __restrict__ msg, const float* __restrict__ heb_in,
               const int* __restrict__ conn, float* __restrict__ heb_out)
{
  const int wid = threadIdx.x >> 5, lane = threadIdx.x & 31;
  const long cell = (long)blockIdx.x * 4 + wid;
  const int b  = (int)(cell / NTOT);
  const int n  = (int)(cell % NTOT);
  const int nc = n >> 8, c = n & 255;
  const float* mb = msg + (long)b * NTOT * DD;

  float self = mb[(long)n * DD + lane];
  float mine = 0.0f;
  for (int k = 0; k < KK; ++k) {
    int idx = conn[((long)nc * CC + c) * KK + k];
    float v = self * mb[((long)nc * CC + idx) * DD + lane];
#pragma unroll
    for (int off = 16; off > 0; off >>= 1) v += __shfl_xor(v, off);
    if (lane == k) mine = v;
  }
  if (lane < KK) {
    long o = cell * KK + lane;
    heb_out[o] = 0.9f * heb_in[o] + 0.1f * (mine * (1.0f / DD));
  }
}

// =====================================================================
// Kernel 5: readout = mean over last ALPHA cells of msg
// =====================================================================
__global__ __launch_bounds__(256)
void k_readout(const float* __restrict__ msg, float* __restrict__ ro)
{
  int t = blockIdx.x * 256 + threadIdx.x;      // < BS*NC*D = 16384
  int b  = t / (NCC * DD);
  int r  = t % (NCC * DD);
  int nc = r / DD, d = r % DD;
  const float* base = msg + (((long)b * NCC + nc) * CC + (CC - AALPHA)) * DD + d;
  float s = 0.0f;
#pragma unroll
  for (int j = 0; j < AALPHA; ++j) s += base[(long)j * DD];
  ro[t] = s * (1.0f / AALPHA);
}

// =====================================================================
extern "C" void kernel_launch(void* const* d_in, const int* in_sizes, int n_in,
                              void* d_out, int out_size, void* d_ws, size_t ws_size,
                              hipStream_t stream)
{
  const float* cc    = (const float*)d_in[0];
  const float* h     = (const float*)d_in[1];
  const float* pmsg  = (const float*)d_in[2];
  const float* dlog  = (const float*)d_in[3];
  const float* prim  = (const float*)d_in[4];
  const float* heb   = (const float*)d_in[5];
  const float* sw1   = (const float*)d_in[6];
  const float* sb1   = (const float*)d_in[7];
  const float* sw2   = (const float*)d_in[8];
  const float* sb2   = (const float*)d_in[9];
  const float* mw1   = (const float*)d_in[10];
  const float* mb1   = (const float*)d_in[11];
  const float* mw2   = (const float*)d_in[12];
  const float* mb2   = (const float*)d_in[13];
  const float* modw1 = (const float*)d_in[14];
  const float* modb1 = (const float*)d_in[15];
  const float* modw2 = (const float*)d_in[16];
  const float* modb2 = (const float*)d_in[17];
  const float* nid   = (const float*)d_in[18];
  const int*   conn  = (const int*)d_in[19];
  const int*   bconn = (const int*)d_in[20];

  float* ws      = (float*)d_ws;
  float* mod_out = ws;                                    // BS*N*57  (~29.9 MB)
  float* recv    = ws + (size_t)BSZ * NTOT * MOD_OUT;     // BS*N*32  (16 MB)

  float* out  = (float*)d_out;
  float* ro   = out;                                      // (8, 2048)
  float* hnew = out + (size_t)BSZ * NCC * DD;             // (8,64,256,32)
  float* msg  = hnew + (size_t)BSZ * NTOT * DD;           // (8,64,256,32)
  float* hebo = msg + (size_t)BSZ * NTOT * DD;            // (8,64,256,16)

  k_mod_mlp<<<NTOT, 32, 0, stream>>>(heb, h, dlog, prim, nid,
                                     modw1, modb1, modw2, modb2, mod_out);
  k_received<<<(BSZ * NTOT) / 4, 128, 0, stream>>>(pmsg, mod_out, conn, bconn, cc, recv);
  k_state_msg<<<(BSZ * NTOT) / (16 * 4), 128, 0, stream>>>(h, recv, prim, nid, mod_out,
                                                           sw1, sb1, sw2, sb2,
                                                           mw1, mb1, mw2, mb2,
                                                           hnew, msg);
  k_hebbian<<<(BSZ * NTOT) / 4, 128, 0, stream>>>(msg, heb, conn, hebo);
  k_readout<<<(BSZ * NCC * DD) / 256, 256, 0, stream>>>(msg, ro);
}